// QLoss_12910671691833
// MI455X (gfx1250) — compile-verified
//
#include <hip/hip_runtime.h>

typedef float v2f __attribute__((ext_vector_type(2)));
typedef float v4f __attribute__((ext_vector_type(4)));
typedef float v8f __attribute__((ext_vector_type(8)));

#define MIN_VAL_F 1e-5f
#define PENALTY_F 100000.0f
#define NBLOCKS 1024
#define NTHREADS 256

__device__ __forceinline__ float qerr_elem(float i, float t) {
  // log diff; fast hw log (v_log_f32). Value is discarded by the select when
  // i < MIN_VAL, so log(0) = -inf in that branch is harmless (no arithmetic on it).
  float ld = __logf(i) - __logf(t);
  float v  = (i > t) ? ld : -ld;
  return (i < MIN_VAL_F) ? (1.0f - i) * PENALTY_F : v;
}

// Wave(32)-wide sum using V_WMMA_F32_16X16X4_F32.
// A (16x4, striped 2 VGPRs/lane): lane L<16 holds A[L][0]=p, A[L][1]=0;
// lane L>=16 holds A[L-16][2]=p, A[L-16][3]=0. B = all ones (4x16).
// D[m][n] = sum_k A[m][k] = p[m] + p[m+16], replicated across columns.
// D layout: VGPR r of lane L<16 = D[r][*] = s[r]; lane L>=16 = s[r+8].
// Sum the 8 accumulator regs, then combine the two half-wave partial sums
// with one xor-shuffle. EXEC is all-1s here (no divergence at call site).
__device__ __forceinline__ float wave_sum_wmma(float p) {
  v2f a; a.x = p;    a.y = 0.0f;
  v2f b; b.x = 1.0f; b.y = 1.0f;
  v8f c = {};
  c = __builtin_amdgcn_wmma_f32_16x16x4_f32(
      /*neg_a=*/false, a, /*neg_b=*/false, b,
      /*c_mod=*/(short)0, c, /*reuse_a=*/false, /*reuse_b=*/false);
  float s = ((c[0] + c[1]) + (c[2] + c[3])) + ((c[4] + c[5]) + (c[6] + c[7]));
  s += __shfl_xor(s, 16, 32);   // lanes 0-15 had s[0..7], lanes 16-31 had s[8..15]
  return s;                     // every lane now holds the full 32-lane sum
}

__global__ __launch_bounds__(NTHREADS)
void qloss_partial(const float* __restrict__ in, const float* __restrict__ tgt,
                   float* __restrict__ part, int n) {
  const int n4 = n >> 2;
  const v4f* __restrict__ in4 = (const v4f*)in;
  const v4f* __restrict__ tg4 = (const v4f*)tgt;
  const int stride = gridDim.x * blockDim.x;

  float acc = 0.0f;
  // Streaming phase: non-temporal b128 loads (working set 256MB > 192MB L2,
  // data is single-use -> NT avoids L2 thrash).
  for (int idx = blockIdx.x * blockDim.x + threadIdx.x; idx < n4; idx += stride) {
    v4f a = __builtin_nontemporal_load(&in4[idx]);
    v4f b = __builtin_nontemporal_load(&tg4[idx]);
    acc += qerr_elem(a.x, b.x);
    acc += qerr_elem(a.y, b.y);
    acc += qerr_elem(a.z, b.z);
    acc += qerr_elem(a.w, b.w);
  }
  // Scalar tail (N = 2^25 is divisible by 4, but stay generic).
  for (int idx = (n4 << 2) + blockIdx.x * blockDim.x + threadIdx.x; idx < n;
       idx += stride)
    acc += qerr_elem(in[idx], tgt[idx]);

  // Wave-level reduction via WMMA (uniform control flow here -> EXEC all 1s).
  float wsum = wave_sum_wmma(acc);

  __shared__ float lds[NTHREADS / 32];
  const int lane = threadIdx.x & 31;
  const int wave = threadIdx.x >> 5;
  if (lane == 0) lds[wave] = wsum;
  __syncthreads();
  if (threadIdx.x == 0) {
    float bsum = 0.0f;
#pragma unroll
    for (int i = 0; i < NTHREADS / 32; ++i) bsum += lds[i];
    part[blockIdx.x] = bsum;   // one deterministic partial per block
  }
}

__global__ __launch_bounds__(NTHREADS)
void qloss_finalize(const float* __restrict__ part, int nparts,
                    float* __restrict__ out, int n) {
  // Final combine in f64: sum magnitude ~3e7 (penalty branch), keep the mean exact.
  double acc = 0.0;
  for (int i = threadIdx.x; i < nparts; i += NTHREADS) acc += (double)part[i];
  __shared__ double sd[NTHREADS];
  sd[threadIdx.x] = acc;
  __syncthreads();
  for (int s = NTHREADS / 2; s > 0; s >>= 1) {
    if (threadIdx.x < s) sd[threadIdx.x] += sd[threadIdx.x + s];
    __syncthreads();
  }
  if (threadIdx.x == 0) out[0] = (float)(sd[0] / (double)n);
}

extern "C" void kernel_launch(void* const* d_in, const int* in_sizes, int n_in,
                              void* d_out, int out_size, void* d_ws, size_t ws_size,
                              hipStream_t stream) {
  const float* in  = (const float*)d_in[0];
  const float* tgt = (const float*)d_in[1];
  float* out  = (float*)d_out;
  float* part = (float*)d_ws;          // NBLOCKS floats of scratch
  const int n = in_sizes[0];

  qloss_partial<<<NBLOCKS, NTHREADS, 0, stream>>>(in, tgt, part, n);
  qloss_finalize<<<1, NTHREADS, 0, stream>>>(part, NBLOCKS, out, n);
}